// NHotEncodingLayer_65369402245699
// MI455X (gfx1250) — compile-verified
//
#include <hip/hip_runtime.h>

// N-hot encoding (per-bag bucket count histogram).
//   out[b, v] += 1 for each value v in bag b; bags defined by sorted `offsets`.
// Strategy:
//   Kernel 1: zero 819.2 MB output with 128-bit non-temporal stores
//             (buffer >> 192MB L2, so stream past L2). Pure HBM-write bound.
//   Kernel 2: 1 thread per value; binary search offsets (staged in LDS) for
//             the segment, then global_atomic_add_f32 (no-return).

typedef __attribute__((ext_vector_type(4))) float v4f;

__global__ void nhot_zero_kernel(float* __restrict__ out, long long n) {
    const long long n4     = n >> 2;                       // # of float4 chunks
    const long long stride = (long long)gridDim.x * blockDim.x;
    v4f  z    = {0.0f, 0.0f, 0.0f, 0.0f};
    v4f* out4 = (v4f*)out;

    for (long long i = (long long)blockIdx.x * blockDim.x + threadIdx.x;
         i < n4; i += stride) {
        __builtin_nontemporal_store(z, out4 + i);          // global_store_b128, TH=NT
    }
    // Tail (n not divisible by 4) — scalar NT stores.
    for (long long i = (n4 << 2) + (long long)blockIdx.x * blockDim.x + threadIdx.x;
         i < n; i += stride) {
        __builtin_nontemporal_store(0.0f, out + i);
    }
}

__global__ void nhot_scatter_kernel(const int* __restrict__ values,
                                    const int* __restrict__ offsets,
                                    const int* __restrict__ num_buckets_p,
                                    float* __restrict__ out,
                                    int total, int nbags) {
    extern __shared__ int s_off[];                         // bag start offsets (8 KB for 2048 bags)
    for (int j = threadIdx.x; j < nbags; j += blockDim.x)
        s_off[j] = offsets[j];
    __syncthreads();

    const int i = blockIdx.x * blockDim.x + threadIdx.x;
    if (i >= total) return;

    const long long N = (long long)num_buckets_p[0];       // device scalar
    const int       v = values[i];

    // seg = upper_bound(offsets, i) - 1   (searchsorted side='right' - 1)
    int lo = 0, hi = nbags;
    while (lo < hi) {
        int mid = (lo + hi) >> 1;
        if (s_off[mid] <= i) lo = mid + 1; else hi = mid;
    }
    const long long seg = (long long)(lo - 1);

    // Counts are small integers -> fp32 accumulation is exact & deterministic.
    atomicAdd(out + seg * N + (long long)v, 1.0f);         // global_atomic_add_f32 (no return)
}

extern "C" void kernel_launch(void* const* d_in, const int* in_sizes, int n_in,
                              void* d_out, int out_size, void* d_ws, size_t ws_size,
                              hipStream_t stream) {
    const int* values  = (const int*)d_in[0];   // (B*L,) int32
    const int* offsets = (const int*)d_in[1];   // (B,)   int32, sorted ascending
    const int* nb_ptr  = (const int*)d_in[2];   // scalar num_buckets (device)
    float*     out     = (float*)d_out;         // (B * num_buckets) fp32

    const long long n     = (long long)out_size;   // 204,800,000 floats (819.2 MB)
    const int       total = in_sizes[0];           // 102,400 values
    const int       nbags = in_sizes[1];           // 2048 bags

    // --- Zero fill: grid-stride over float4 chunks, saturate HBM writes ---
    const int zthreads = 256;
    long long n4 = (n + 3) >> 2;
    long long zb = (n4 + zthreads - 1) / zthreads;
    if (zb > 16384) zb = 16384;                    // grid-stride: ~12 B128 stores/thread
    if (zb < 1)     zb = 1;
    nhot_zero_kernel<<<dim3((unsigned)zb), dim3(zthreads), 0, stream>>>(out, n);

    // --- Scatter: one thread per value ---
    const int sthreads = 256;
    const int sblocks  = (total + sthreads - 1) / sthreads;
    const size_t shmem = (size_t)nbags * sizeof(int);
    nhot_scatter_kernel<<<dim3(sblocks), dim3(sthreads), shmem, stream>>>(
        values, offsets, nb_ptr, out, total, nbags);
}